// EdgeAwareGCN_28312424415404
// MI455X (gfx1250) — compile-verified
//
#include <hip/hip_runtime.h>
#include <hip/hip_bf16.h>

typedef __attribute__((ext_vector_type(2))) float v2f;
typedef __attribute__((ext_vector_type(8))) float v8f;

#define N_NODES  50000
#define N_EDGES  800000
#define NODE_DIM 128
#define EDGE_DIM 64
#define HIDDEN   128
#define OUT_DIM  64

// -------------------------------------------------------------------------
// no-return f32 atomic add -> global_atomic_add_f32 (STOREcnt, L2 atomic unit)
__device__ __forceinline__ void atomAddF(float* p, float v) {
    __hip_atomic_fetch_add(p, v, __ATOMIC_RELAXED, __HIP_MEMORY_SCOPE_AGENT);
}

// -------------------------------------------------------------------------
// Tiled WMMA fp32 GEMM:  Y[r][c] = sum_k A[r][k] * W[k][c] + bias[c]
//   block = 256 threads (8 waves), tile = 128 rows x 16 cols, K chunked by 64.
//   MODE 0: A = A0
//   MODE 1: A = relu(A0)                                   (fused activation)
//   MODE 2: A = concat(relu(A0) + A1, A2)  halves of K/2   (fused concat+add)
// Uses V_WMMA_F32_16X16X4_F32; fragment layouts per ISA 7.12.2:
//   A 16x4 : lane L -> row M=L&15, K = (L>=16?2:0)+{0,1}  (v2f)
//   B 4x16 : lane L -> col N=L&15, K = (L>=16?2:0)+{0,1}  (v2f, LDS transposed)
//   D 16x16: lane L, vgpr v -> row M = v + (L>=16?8:0), col N = L&15
template<int K, int MODE>
__global__ __launch_bounds__(256) void gemm_wmma_f32(
    const float* __restrict__ A0, const float* __restrict__ A1,
    const float* __restrict__ A2,
    const float* __restrict__ W,  const float* __restrict__ bias,
    float* __restrict__ Y, int nrows, int H)
{
    constexpr int KC  = 64;       // K chunk staged in LDS
    constexpr int LDA = KC + 4;   // pad: row stride 68 dwords -> banks 4 apart
    constexpr int BM  = 128;

    __shared__ float ldsA[BM * LDA];   // 34.0 KB
    __shared__ float ldsB[16 * LDA];   //  4.3 KB  (transposed: [N][K])

    const int t       = threadIdx.x;
    const int rowBase = blockIdx.x * BM;
    const int cb      = blockIdx.y * 16;
    const int wave    = t >> 5;
    const int lane    = t & 31;
    const int hi2     = (lane >> 4) << 1;   // 0 or 2
    const int m16     = lane & 15;

    v8f acc = {};

    for (int kc = 0; kc < K; kc += KC) {
        // ---- stage A chunk: 128 rows x 64 k (2048 float4, 8 per thread) ----
        #pragma unroll
        for (int idx = t; idx < BM * (KC / 4); idx += 256) {
            const int r  = idx / (KC / 4);
            const int kk = (idx % (KC / 4)) * 4;
            const int gk = kc + kk;
            const int gr = rowBase + r;
            float4 v = make_float4(0.f, 0.f, 0.f, 0.f);
            if (gr < nrows) {
                if (MODE == 0) {
                    v = *(const float4*)&A0[(size_t)gr * K + gk];
                } else if (MODE == 1) {
                    float4 u = *(const float4*)&A0[(size_t)gr * K + gk];
                    v.x = fmaxf(u.x, 0.f); v.y = fmaxf(u.y, 0.f);
                    v.z = fmaxf(u.z, 0.f); v.w = fmaxf(u.w, 0.f);
                } else { // MODE 2: concat(relu(A0)+A1, A2), halves of K/2
                    if (gk < K / 2) {
                        float4 u = *(const float4*)&A0[(size_t)gr * (K / 2) + gk];
                        float4 w = *(const float4*)&A1[(size_t)gr * (K / 2) + gk];
                        v.x = fmaxf(u.x, 0.f) + w.x; v.y = fmaxf(u.y, 0.f) + w.y;
                        v.z = fmaxf(u.z, 0.f) + w.z; v.w = fmaxf(u.w, 0.f) + w.w;
                    } else {
                        v = *(const float4*)&A2[(size_t)gr * (K / 2) + (gk - K / 2)];
                    }
                }
            }
            *(float4*)&ldsA[r * LDA + kk] = v;
        }
        // ---- stage B chunk transposed: ldsB[n*LDA + k], 64 k x 16 n --------
        {
            const int kr = t >> 2;          // 0..63
            const int c  = (t & 3) * 4;     // 0,4,8,12
            const float4 wv = *(const float4*)&W[(size_t)(kc + kr) * H + cb + c];
            ldsB[(c + 0) * LDA + kr] = wv.x;
            ldsB[(c + 1) * LDA + kr] = wv.y;
            ldsB[(c + 2) * LDA + kr] = wv.z;
            ldsB[(c + 3) * LDA + kr] = wv.w;
        }
        __syncthreads();

        const float* ap = &ldsA[(wave * 16 + m16) * LDA + hi2];
        const float* bp = &ldsB[m16 * LDA + hi2];
        #pragma unroll
        for (int k0 = 0; k0 < KC; k0 += 4) {
            v2f a = *(const v2f*)(ap + k0);   // 8B-aligned ds_load_b64
            v2f b = *(const v2f*)(bp + k0);
            acc = __builtin_amdgcn_wmma_f32_16x16x4_f32(
                false, a, false, b, (short)0, acc, false, false);
        }
        __syncthreads();
    }

    // ---- epilogue: bias + store ----
    const float bv  = bias[cb + m16];
    const int   col = cb + m16;
    #pragma unroll
    for (int v = 0; v < 8; v++) {
        const int gr = rowBase + wave * 16 + v + hi2 * 4;  // + (hi?8:0)
        if (gr < nrows)
            Y[(size_t)gr * H + col] = acc[v] + bv;
    }
}

// -------------------------------------------------------------------------
// Scatter-add of 128-float rows: out[dst[e]][:] += (RELU? relu:id)(in[s][:])
// s = e (SELF) or src[e].  One wave per edge: lane owns floats 4l..4l+3
// -> one global_load_b128 per lane (512B coalesced) + 4 f32 atomics.
template<bool RELU, bool SELF>
__global__ __launch_bounds__(256) void scatter_add_rows(
    const float* __restrict__ in, const int* __restrict__ src,
    const int* __restrict__ dst, float* __restrict__ out, int nEdges)
{
    const int wave = threadIdx.x >> 5;
    const int lane = threadIdx.x & 31;
    const int e    = blockIdx.x * 8 + wave;
    if (e >= nEdges) return;

    const int s = SELF ? e : src[e];
    const int d = dst[e];

    float4 v = *(const float4*)(in + (size_t)s * HIDDEN + lane * 4);
    if (RELU) {
        v.x = fmaxf(v.x, 0.f); v.y = fmaxf(v.y, 0.f);
        v.z = fmaxf(v.z, 0.f); v.w = fmaxf(v.w, 0.f);
    }
    float* po = out + (size_t)d * HIDDEN + lane * 4;
    atomAddF(po + 0, v.x);
    atomAddF(po + 1, v.y);
    atomAddF(po + 2, v.z);
    atomAddF(po + 3, v.w);
}

// -------------------------------------------------------------------------
extern "C" void kernel_launch(void* const* d_in, const int* in_sizes, int n_in,
                              void* d_out, int out_size, void* d_ws, size_t ws_size,
                              hipStream_t stream) {
    (void)in_sizes; (void)n_in; (void)out_size; (void)ws_size;

    const float* node_x = (const float*)d_in[0];
    const int*   eidx   = (const int*)  d_in[1];   // [2, N_EDGES]
    const float* edge_x = (const float*)d_in[2];
    const int*   emidx  = (const int*)  d_in[3];   // [2, N_EDGES]
    const float* W1  = (const float*)d_in[4];  const float* b1  = (const float*)d_in[5];
    const float* W2  = (const float*)d_in[6];  const float* b2  = (const float*)d_in[7];
    const float* We  = (const float*)d_in[8];  const float* be  = (const float*)d_in[9];
    const float* Wfc = (const float*)d_in[10]; const float* bfc = (const float*)d_in[11];
    float* out = (float*)d_out;

    const int* row  = eidx;                const int* col  = eidx  + N_EDGES;
    const int* mrow = emidx;               const int* mcol = emidx + N_EDGES;

    // ---- workspace carve-up ----
    const size_t NODE_F = (size_t)N_NODES * HIDDEN * sizeof(float);  // 25.6 MB
    const size_t EDGE_F = (size_t)N_EDGES * HIDDEN * sizeof(float);  // 409.6 MB
    char*  ws      = (char*)d_ws;
    float* bufH    = (float*)ws;            ws += NODE_F;  // h1 then h2
    float* bufX1   = (float*)ws;            ws += NODE_F;  // agg1 (pre-relu)
    float* bufX2   = (float*)ws;            ws += NODE_F;  // agg2 (pre-relu)
    float* bufAggN = (float*)ws;            ws += NODE_F;
    float* bufAggE = (float*)ws;            ws += NODE_F;
    float* bufHE   = (float*)ws;            ws += EDGE_F;  // edge_x @ We + be
    float* bufE    = (float*)ws;            ws += EDGE_F;  // line-graph agg

    // zero the atomic accumulators (graph-capturable async memset)
    hipMemsetAsync(bufX1,   0, NODE_F, stream);
    hipMemsetAsync(bufX2,   0, NODE_F, stream);
    hipMemsetAsync(bufAggN, 0, NODE_F, stream);
    hipMemsetAsync(bufAggE, 0, NODE_F, stream);
    hipMemsetAsync(bufE,    0, EDGE_F, stream);

    const dim3 blk(256);
    const dim3 gNode((N_NODES + 127) / 128, HIDDEN / 16);
    const dim3 gEdge( N_EDGES / 128,        HIDDEN / 16);
    const dim3 gOut ((N_NODES + 127) / 128, OUT_DIM / 16);
    const int  sBlocks = (N_EDGES + 7) / 8;

    // layer 1: h = node_x @ W1 + b1 ; x1_agg[col] += h[row]
    gemm_wmma_f32<128, 0><<<gNode, blk, 0, stream>>>(node_x, nullptr, nullptr,
                                                     W1, b1, bufH, N_NODES, HIDDEN);
    scatter_add_rows<false, false><<<sBlocks, blk, 0, stream>>>(bufH, row, col, bufX1, N_EDGES);

    // layer 2: h = relu(x1_agg) @ W2 + b2 ; x2_agg[col] += h[row]
    gemm_wmma_f32<128, 1><<<gNode, blk, 0, stream>>>(bufX1, nullptr, nullptr,
                                                     W2, b2, bufH, N_NODES, HIDDEN);
    scatter_add_rows<false, false><<<sBlocks, blk, 0, stream>>>(bufH, row, col, bufX2, N_EDGES);

    // edge conv on line graph: he = edge_x @ We + be ; e[mcol] += he[mrow]
    gemm_wmma_f32<64, 0><<<gEdge, blk, 0, stream>>>(edge_x, nullptr, nullptr,
                                                    We, be, bufHE, N_EDGES, HIDDEN);
    scatter_add_rows<false, false><<<sBlocks, blk, 0, stream>>>(bufHE, mrow, mcol, bufE, N_EDGES);

    // aggregations: aggN[col] += relu(x2_agg)[row] ; aggE[col] += e[eid]
    scatter_add_rows<true,  false><<<sBlocks, blk, 0, stream>>>(bufX2, row, col, bufAggN, N_EDGES);
    scatter_add_rows<false, true ><<<sBlocks, blk, 0, stream>>>(bufE, nullptr, col, bufAggE, N_EDGES);

    // out = concat(relu(x2_agg)+aggN, aggE) @ Wfc + bfc   (concat fused in A-loader)
    gemm_wmma_f32<256, 2><<<gOut, blk, 0, stream>>>(bufX2, bufAggN, bufAggE,
                                                    Wfc, bfc, out, N_NODES, OUT_DIM);
}